// DepthwiseCausalConv1d_57870389346354
// MI455X (gfx1250) — compile-verified
//
#include <hip/hip_runtime.h>
#include <stdint.h>

// ---------------------------------------------------------------------------
// Depthwise causal conv1d, B=8 T=4096 C=1024 K=4, dilation=1.
//   y[b,t,c] = w[c,3]*x[t] + w[c,2]*x[t-1] + w[c,1]*x[t-2] + w[c,0]*x[t-3]
// AI ~ 1 FLOP/byte -> HBM-bound (268 MB @ 23.3 TB/s ~ 11.5 us floor).
// Data path: TDM (tensor_load_to_lds, non-temporal) global->LDS double-buffered
// tiles, register sliding window over t, non-temporal b128 stores.
// ---------------------------------------------------------------------------

typedef uint32_t v4u __attribute__((ext_vector_type(4)));
typedef uint32_t v8u __attribute__((ext_vector_type(8)));
typedef float    v4f __attribute__((ext_vector_type(4)));

constexpr int Bn = 8, Tn = 4096, Cn = 1024, Kn = 4;
constexpr int TT      = 8;              // t-rows per TDM tile
constexpr int NTILES  = 16;             // tiles per block
constexpr int TCHUNK  = TT * NTILES;    // 128 t-rows per block
constexpr int THREADS = Cn / 4;         // 256 threads: 4 channels/thread
constexpr int SLOT_FLOATS  = TT * Cn;   // 8192 floats = 32 KB per slot
constexpr size_t SMEM_BYTES = 2ull * SLOT_FLOATS * sizeof(float); // 64 KB

// Issue one TDM 2D tile load: (TT rows) x (Cn floats), row stride Cn floats,
// from global address `gsrc` into LDS byte offset `lds_byte_addr`.
__device__ __forceinline__ void tdm_issue_tile(const float* gsrc,
                                               uint32_t lds_byte_addr) {
  const uint64_t ga = (uint64_t)(uintptr_t)gsrc;

  // D# group 0 (4 SGPRs)
  v4u g0;
  g0[0] = 1u;                                                // count=1, user mode
  g0[1] = lds_byte_addr;                                     // lds_addr (bytes)
  g0[2] = (uint32_t)ga;                                      // global_addr[31:0]
  g0[3] = ((uint32_t)(ga >> 32) & 0x01FFFFFFu)               // global_addr[56:32]
          | 0x80000000u;                                     // type=2 ("image")

  // D# group 1 (8 SGPRs)
  v8u g1;
  g1[0] = (2u << 16);                                        // data_size=4B; no mask/pad/iter
  g1[1] = ((uint32_t)Cn & 0xFFFFu) << 16;                    // tensor_dim0[15:0]
  g1[2] = ((uint32_t)Cn >> 16) | (((uint32_t)TT & 0xFFFFu) << 16); // dim0[31:16]|dim1[15:0]
  g1[3] = (((uint32_t)TT >> 16) & 0xFFFFu) | (((uint32_t)Cn & 0xFFFFu) << 16); // dim1 hi|tile_dim0
  g1[4] = (uint32_t)TT;                                      // tile_dim1 (tile_dim2=0)
  g1[5] = (uint32_t)Cn;                                      // tensor_dim0_stride[31:0]
  g1[6] = 0u;
  g1[7] = 0u;

  // Read-once stream: non-temporal load policy.
  asm volatile("tensor_load_to_lds %0, %1 th:TH_LOAD_NT" :: "s"(g0), "s"(g1) : "memory");
}

__global__ __launch_bounds__(THREADS)
void dwconv_tdm_kernel(const float* __restrict__ x,
                       const float* __restrict__ w,
                       float* __restrict__ y) {
  extern __shared__ float smem[];                 // 2 slots x 32 KB
  const int tid = threadIdx.x;
  const int c   = tid * 4;

  const int blocksPerB = Tn / TCHUNK;             // 32
  const int b  = blockIdx.x / blocksPerB;
  const int t0 = (blockIdx.x - b * blocksPerB) * TCHUNK;

  // Per-thread taps for 4 channels: weight is (C,K) row-major.
  const v4f wv0 = *(const v4f*)(w + (size_t)(c + 0) * Kn);
  const v4f wv1 = *(const v4f*)(w + (size_t)(c + 1) * Kn);
  const v4f wv2 = *(const v4f*)(w + (size_t)(c + 2) * Kn);
  const v4f wv3 = *(const v4f*)(w + (size_t)(c + 3) * Kn);

  const float* xb = x + (size_t)b * Tn * Cn;

  // Branchless causal history: clamp row to >=0 (legal address) and scale by
  // a uniform 0/1 mask. One b128 load per history row, no divergent guards.
  const int   r1 = (t0 >= 1) ? t0 - 1 : 0;
  const int   r2 = (t0 >= 2) ? t0 - 2 : 0;
  const int   r3 = (t0 >= 3) ? t0 - 3 : 0;
  const float m1 = (t0 >= 1) ? 1.f : 0.f;
  const float m2 = (t0 >= 2) ? 1.f : 0.f;
  const float m3 = (t0 >= 3) ? 1.f : 0.f;
  v4f p1 = *(const v4f*)(xb + (size_t)r1 * Cn + c) * m1;
  v4f p2 = *(const v4f*)(xb + (size_t)r2 * Cn + c) * m2;
  v4f p3 = *(const v4f*)(xb + (size_t)r3 * Cn + c) * m3;

  // Low 32 bits of a flat LDS pointer are the LDS byte offset.
  const uint32_t lds_base = (uint32_t)(uintptr_t)smem;

  // Wave-uniform lead-wave predicate (TDM ignores EXEC; must scalar-branch).
  const int wave_id = __builtin_amdgcn_readfirstlane(tid >> 5);
  const bool lead = (wave_id == 0);

  if (lead) {
    tdm_issue_tile(xb + (size_t)t0 * Cn, lds_base);          // prefetch tile 0
  }

  for (int tile = 0; tile < NTILES; ++tile) {
    if (lead) {
      if (tile + 1 < NTILES) {
        // Prefetch tile+1 into the other slot (its previous readers finished
        // at the trailing barrier of iteration tile-1), then wait so the
        // oldest outstanding TDM op (tile) is complete: in-order completion.
        tdm_issue_tile(xb + (size_t)(t0 + (tile + 1) * TT) * Cn,
                       lds_base + (uint32_t)(((tile + 1) & 1) * SLOT_FLOATS * 4));
        __builtin_amdgcn_s_wait_tensorcnt(1);
      } else {
        __builtin_amdgcn_s_wait_tensorcnt(0);
      }
    }
    __syncthreads();                               // publish tile to all waves

    const float* buf  = smem + (size_t)(tile & 1) * SLOT_FLOATS;
    float*       yrow = y + ((size_t)b * Tn + (size_t)(t0 + tile * TT)) * Cn + c;

#pragma unroll
    for (int r = 0; r < TT; ++r) {
      const v4f cur = *(const v4f*)(buf + (size_t)r * Cn + c); // ds_load_b128
      v4f o;
      o.x = wv0.w * cur.x + wv0.z * p1.x + wv0.y * p2.x + wv0.x * p3.x;
      o.y = wv1.w * cur.y + wv1.z * p1.y + wv1.y * p2.y + wv1.x * p3.y;
      o.z = wv2.w * cur.z + wv2.z * p1.z + wv2.y * p2.z + wv2.x * p3.z;
      o.w = wv3.w * cur.w + wv3.z * p1.w + wv3.y * p2.w + wv3.x * p3.w;
      // Write-once stream: non-temporal b128 store (th:TH_STORE_NT).
      __builtin_nontemporal_store(o, (v4f*)(yrow + (size_t)r * Cn));
      p3 = p2; p2 = p1; p1 = cur;                  // causal sliding window
    }
    __syncthreads();                               // slot free for tile+2
  }
}

extern "C" void kernel_launch(void* const* d_in, const int* in_sizes, int n_in,
                              void* d_out, int out_size, void* d_ws, size_t ws_size,
                              hipStream_t stream) {
  (void)in_sizes; (void)n_in; (void)out_size; (void)d_ws; (void)ws_size;
  const float* x = (const float*)d_in[0];   // (B, T, C) f32
  const float* w = (const float*)d_in[1];   // (C, K)    f32
  float*       y = (float*)d_out;           // (B, T, C) f32

  dim3 grid(Bn * (Tn / TCHUNK));            // 256 blocks
  dim3 block(THREADS);                      // 256 threads = 8 wave32
  dwconv_tdm_kernel<<<grid, block, SMEM_BYTES, stream>>>(x, w, y);
}